// AngProtoLossStable_11527692223246
// MI455X (gfx1250) — compile-verified
//
#include <hip/hip_runtime.h>
#include <math.h>

typedef __attribute__((ext_vector_type(2))) float v2f;
typedef __attribute__((ext_vector_type(8))) float v8f;

#define NN 4096
#define MM 16
#define DD 512
#define ROWS_PER_BLK 32   // two 16-row WMMA panels per block

// ---------------- Phase A: centroids, u, inverse norms ----------------
__global__ __launch_bounds__(256)
void centroid_kernel(const float* __restrict__ dvecs,
                     float* __restrict__ u,
                     float* __restrict__ cent,
                     float* __restrict__ inv_nu,
                     float* __restrict__ inv_nc) {
    __shared__ float red_u[256];
    __shared__ float red_c[256];
    const int n = blockIdx.x;
    const int t = threadIdx.x;
    const float* base = dvecs + (size_t)n * MM * DD;
    float su = 0.f, sc = 0.f;
    for (int d = t; d < DD; d += 256) {
        float acc = 0.f;
        #pragma unroll
        for (int m = 0; m < MM; ++m) acc += base[m * DD + d];
        const float c  = acc * (1.0f / 16.0f);
        const float uu = base[15 * DD + d];
        cent[(size_t)n * DD + d] = c;
        u   [(size_t)n * DD + d] = uu;
        su += uu * uu;
        sc += c * c;
    }
    red_u[t] = su; red_c[t] = sc;
    __syncthreads();
    for (int s = 128; s > 0; s >>= 1) {
        if (t < s) { red_u[t] += red_u[t + s]; red_c[t] += red_c[t + s]; }
        __syncthreads();
    }
    if (t == 0) {
        inv_nu[n] = rsqrtf(red_u[0]);
        inv_nc[n] = rsqrtf(red_c[0]);
    }
}

// ---------------- Phase B+C: WMMA GEMM + fused online log-softmax ----------------
// Block b: rows [32b, 32b+32) as two 16-row A panels staged in LDS.
// 8 waves; wave w sweeps its 32 column tiles in PAIRS (N-block x2).
// K loop: software-pipelined chunks of 16 (4 wmma steps); per step:
// 2 ds_load_b64 (A panels) + 2 global_load_b64 (B tiles) -> 4 WMMAs.
__global__ __launch_bounds__(256)
void gemm_softmax_kernel(const float* __restrict__ u,
                         const float* __restrict__ cent,
                         const float* __restrict__ inv_nu,
                         const float* __restrict__ inv_nc,
                         const float* __restrict__ wp,
                         const float* __restrict__ bp,
                         float* __restrict__ out) {
    __shared__ float aLds[2][16][DD];              // 64 KB: two A panels
    __shared__ float nu_l[ROWS_PER_BLK];
    __shared__ float diag_l[ROWS_PER_BLK];
    __shared__ float part_max[ROWS_PER_BLK][128];  // 16 KB
    __shared__ float part_sum[ROWS_PER_BLK][128];  // 16 KB
    __shared__ float red_max[ROWS_PER_BLK][8];
    __shared__ float red_sum[ROWS_PER_BLK][8];

    const int t        = threadIdx.x;
    const int row_base = blockIdx.x * ROWS_PER_BLK;
    const float w = wp[0];
    const float b = bp[0];
    if (t < ROWS_PER_BLK) nu_l[t] = inv_nu[row_base + t];

    // stage A: rows row_base..row_base+31 are contiguous in u
    {
        const float4* src = (const float4*)(u + (size_t)row_base * DD);
        float4* dst = (float4*)(&aLds[0][0][0]);
        #pragma unroll
        for (int i = 0; i < (ROWS_PER_BLK * DD / 4) / 256; ++i)
            dst[t + i * 256] = src[t + i * 256];
    }
    __syncthreads();

    const int wave = t >> 5;
    const int lane = t & 31;
    const int h    = lane >> 4;   // lane half: k += 2h
    const int i16  = lane & 15;
    const int k2h  = 2 * h;

    // per-lane online softmax state: slot p*8+r -> block row p*16 + r + 8h
    float rmax[16], rsum[16];
    #pragma unroll
    for (int r = 0; r < 16; ++r) { rmax[r] = -1e30f; rsum[r] = 0.f; }

    #pragma unroll 1
    for (int cti = 0; cti < 16; ++cti) {              // 2 col tiles per iter
        const int ct0 = wave * 32 + cti * 2;
        const float* brow0 = cent + (size_t)(ct0 * 16 + i16) * DD;
        const float* brow1 = cent + (size_t)((ct0 + 1) * 16 + i16) * DD;

        v8f acc[2][2] = {{v8f{}, v8f{}}, {v8f{}, v8f{}}};  // [tile][panel]
        v2f b0n[4], b1n[4];
        #pragma unroll
        for (int j = 0; j < 4; ++j) {                 // prologue: chunk 0
            b0n[j] = *(const v2f*)(brow0 + j * 4 + k2h);
            b1n[j] = *(const v2f*)(brow1 + j * 4 + k2h);
        }

        #pragma unroll 1
        for (int kc = 0; kc < DD - 16; kc += 16) {    // steady state
            v2f b0c[4], b1c[4];
            #pragma unroll
            for (int j = 0; j < 4; ++j) { b0c[j] = b0n[j]; b1c[j] = b1n[j]; }
            #pragma unroll
            for (int j = 0; j < 4; ++j) {             // prefetch chunk kc+16
                b0n[j] = *(const v2f*)(brow0 + kc + 16 + j * 4 + k2h);
                b1n[j] = *(const v2f*)(brow1 + kc + 16 + j * 4 + k2h);
            }
            #pragma unroll
            for (int j = 0; j < 4; ++j) {             // compute chunk kc
                const int k = kc + j * 4;
                v2f a0 = *(const v2f*)(&aLds[0][i16][k + k2h]);
                v2f a1 = *(const v2f*)(&aLds[1][i16][k + k2h]);
                acc[0][0] = __builtin_amdgcn_wmma_f32_16x16x4_f32(
                    false, a0, false, b0c[j], (short)0, acc[0][0], false, false);
                acc[0][1] = __builtin_amdgcn_wmma_f32_16x16x4_f32(
                    false, a1, false, b0c[j], (short)0, acc[0][1], false, false);
                acc[1][0] = __builtin_amdgcn_wmma_f32_16x16x4_f32(
                    false, a0, false, b1c[j], (short)0, acc[1][0], false, false);
                acc[1][1] = __builtin_amdgcn_wmma_f32_16x16x4_f32(
                    false, a1, false, b1c[j], (short)0, acc[1][1], false, false);
            }
        }
        #pragma unroll
        for (int j = 0; j < 4; ++j) {                 // epilogue: last chunk
            const int k = (DD - 16) + j * 4;
            v2f a0 = *(const v2f*)(&aLds[0][i16][k + k2h]);
            v2f a1 = *(const v2f*)(&aLds[1][i16][k + k2h]);
            acc[0][0] = __builtin_amdgcn_wmma_f32_16x16x4_f32(
                false, a0, false, b0n[j], (short)0, acc[0][0], false, false);
            acc[0][1] = __builtin_amdgcn_wmma_f32_16x16x4_f32(
                false, a1, false, b0n[j], (short)0, acc[0][1], false, false);
            acc[1][0] = __builtin_amdgcn_wmma_f32_16x16x4_f32(
                false, a0, false, b1n[j], (short)0, acc[1][0], false, false);
            acc[1][1] = __builtin_amdgcn_wmma_f32_16x16x4_f32(
                false, a1, false, b1n[j], (short)0, acc[1][1], false, false);
        }

        // scale -> clip -> logits -> online softmax / diagonal capture
        #pragma unroll
        for (int ti = 0; ti < 2; ++ti) {
            const int cb = (ct0 + ti) * 16;
            const int gn = cb + i16;
            const float inc = inv_nc[gn];
            #pragma unroll
            for (int p = 0; p < 2; ++p) {
                const bool dt = (cb == row_base + p * 16);
                #pragma unroll
                for (int r = 0; r < 8; ++r) {
                    const int m = r + 8 * h;           // row within panel
                    float cosv  = acc[ti][p][r] * nu_l[p * 16 + m] * inc;
                    cosv        = fmaxf(cosv, 1e-6f);
                    const float lv = cosv * w + b;
                    if (dt && (i16 == m)) diag_l[p * 16 + m] = lv;
                    const int s = p * 8 + r;
                    if (lv > rmax[s]) {
                        rsum[s] = rsum[s] * __expf(rmax[s] - lv) + 1.0f;
                        rmax[s] = lv;
                    } else {
                        rsum[s] += __expf(lv - rmax[s]);
                    }
                }
            }
        }
    }

    // dump per-lane partials: each block row gets 128 (8 waves x 16 lanes)
    #pragma unroll
    for (int p = 0; p < 2; ++p)
        #pragma unroll
        for (int r = 0; r < 8; ++r) {
            const int m = p * 16 + r + 8 * h;
            part_max[m][wave * 16 + i16] = rmax[p * 8 + r];
            part_sum[m][wave * 16 + i16] = rsum[p * 8 + r];
        }
    __syncthreads();

    // reduce 128 -> 8 -> 1 per row (32 rows x 8 subs = 256 threads)
    const int row = t >> 3;
    const int sub = t & 7;
    {
        float m1 = -1e30f, s1 = 0.f;
        #pragma unroll
        for (int i = 0; i < 16; ++i) {
            const float m2 = part_max[row][sub * 16 + i];
            const float s2 = part_sum[row][sub * 16 + i];
            const float mn = fmaxf(m1, m2);
            s1 = s1 * __expf(m1 - mn) + s2 * __expf(m2 - mn);
            m1 = mn;
        }
        red_max[row][sub] = m1;
        red_sum[row][sub] = s1;
    }
    __syncthreads();
    if (sub == 0) {
        float m1 = -1e30f, s1 = 0.f;
        #pragma unroll
        for (int i = 0; i < 8; ++i) {
            const float m2 = red_max[row][i];
            const float s2 = red_sum[row][i];
            const float mn = fmaxf(m1, m2);
            s1 = s1 * __expf(m1 - mn) + s2 * __expf(m2 - mn);
            m1 = mn;
        }
        // -logp[n,n] = -(diag - max - log(sumexp))
        const float contrib = -(diag_l[row] - m1 - __logf(s1));
        atomicAdd(out, contrib * (1.0f / (float)NN));
    }
}

extern "C" void kernel_launch(void* const* d_in, const int* in_sizes, int n_in,
                              void* d_out, int out_size, void* d_ws, size_t ws_size,
                              hipStream_t stream) {
    const float* dvecs = (const float*)d_in[0];
    const float* wp    = (const float*)d_in[1];
    const float* bp    = (const float*)d_in[2];
    float* out = (float*)d_out;

    // workspace layout (floats): u[N*D] | cent[N*D] | inv_nu[N] | inv_nc[N]
    float* ws_u   = (float*)d_ws;
    float* ws_c   = ws_u + (size_t)NN * DD;
    float* inv_nu = ws_c + (size_t)NN * DD;
    float* inv_nc = inv_nu + NN;

    hipMemsetAsync(d_out, 0, sizeof(float), stream);

    centroid_kernel<<<NN, 256, 0, stream>>>(dvecs, ws_u, ws_c, inv_nu, inv_nc);
    gemm_softmax_kernel<<<NN / ROWS_PER_BLK, 256, 0, stream>>>(
        ws_u, ws_c, inv_nu, inv_nc, wp, bp, out);
}